// Autoencoder_35940286333260
// MI455X (gfx1250) — compile-verified
//
#include <hip/hip_runtime.h>
#include <math.h>

// ---------------- problem constants ----------------
constexpr int N_NODES = 50000;               // multiple of 16 -> no row guards in GEMM
constexpr int E_EDGES = 800000;
constexpr int EP      = E_EDGES + N_NODES;   // edges incl. self loops
constexpr int XS      = 128;                 // padded row stride (floats) for x/h/out
constexpr float NEG_SLOPE = 0.2f;

typedef float v2f __attribute__((ext_vector_type(2)));
typedef float v8f __attribute__((ext_vector_type(8)));

static inline int cdiv(int a, int b) { return (a + b - 1) / b; }

// ---------------- helpers ----------------
__device__ __forceinline__ float leaky(float v) {
    return v > 0.0f ? v : NEG_SLOPE * v;
}

// sign-aware float atomic max via integer atomics (valid for all finite floats / -inf init)
__device__ __forceinline__ void atomicMaxF32(float* addr, float val) {
    if (val >= 0.0f)
        atomicMax(reinterpret_cast<int*>(addr), __float_as_int(val));
    else
        atomicMin(reinterpret_cast<unsigned int*>(addr), __float_as_uint(val));
}

__device__ __forceinline__ void edge_nodes(const int* __restrict__ ei, int e, int& s, int& d) {
    if (e < E_EDGES) { s = ei[e]; d = ei[E_EDGES + e]; }
    else             { s = e - E_EDGES; d = s; }   // appended self loop
}

// ---------------- kernel 0: build padded input x = [emb[id] | feats | 0-pad] ----------------
__global__ void build_x(const int* __restrict__ ids, const float* __restrict__ feats,
                        const float* __restrict__ emb, float* __restrict__ xb) {
    int t = blockIdx.x * blockDim.x + threadIdx.x;
    int n = t >> 7, j = t & 127;
    if (n >= N_NODES) return;
    float v = 0.0f;
    if (j < 8)        v = emb[ids[n] * 8 + j];
    else if (j < 23)  v = feats[n * 15 + (j - 8)];
    xb[n * XS + j] = v;
}

// ---------------- kernel 1a: pre-swizzle W into per-lane B fragments ----------------
// Wpk layout: [tile = n/16][kb = k/4][lane 0..31][2], where for lane L (half=L>>4, col=L&15):
//   element j holds W[kb*4 + half*2 + j][tile*16 + col]  (zero beyond Kreal).
// Makes the GEMM B-load a single contiguous b64 per lane per K-step.
__global__ void pack_w(const float* __restrict__ W, float* __restrict__ Wpk,
                       int Kreal, int Fout, int KB) {
    int idx = blockIdx.x * blockDim.x + threadIdx.x;
    int total = (Fout / 16) * KB * 64;
    if (idx >= total) return;
    int j   = idx & 1;
    int L   = (idx >> 1) & 31;
    int rem = idx >> 6;
    int kb  = rem % KB;
    int t   = rem / KB;
    int k   = kb * 4 + (L >> 4) * 2 + j;
    int n   = t * 16 + (L & 15);
    Wpk[idx] = (k < Kreal) ? W[k * Fout + n] : 0.0f;
}

// ---------------- kernel 1b: WMMA GEMM  h[16x16 tile] = x @ W ----------------
// One wave per tile. Branch-free inner loop: 2x global_load_b64 + 1x v_wmma_f32_16x16x4_f32.
__global__ __launch_bounds__(32) void gat_gemm(const float* __restrict__ x,
                                               const float* __restrict__ Wpk,
                                               float* __restrict__ h,
                                               int KB) {
    const int m0   = blockIdx.x * 16;
    const int tile = blockIdx.y;
    const int lane = threadIdx.x;      // 0..31 (wave32)
    const int half = lane >> 4;        // A: K-pair select; C/D: row-half select
    const int l15  = lane & 15;

    // A 16x4 f32 (ISA 7.12.2): lane row = m0+l15, VGPRs hold K = {k0,k0+1}+half*2
    const v2f* __restrict__ xp = (const v2f*)(x + (size_t)(m0 + l15) * XS);
    // B fragments pre-swizzled: consecutive v2f per lane, 32 v2f per K-step
    const v2f* __restrict__ bp = (const v2f*)(Wpk + (size_t)tile * KB * 64) + lane;

    v8f c = {0.f, 0.f, 0.f, 0.f, 0.f, 0.f, 0.f, 0.f};
#pragma unroll 4
    for (int kb = 0; kb < KB; ++kb) {
        v2f a = xp[kb * 2 + half];
        v2f b = bp[kb * 32];
        c = __builtin_amdgcn_wmma_f32_16x16x4_f32(false, a, false, b,
                                                  (short)0, c, false, false);
    }

    // C/D: VGPR r -> row m0 + r + half*8, col = tile*16 + l15 (no guard: N % 16 == 0)
    float* hp = h + (size_t)(m0 + half * 8) * XS + tile * 16 + l15;
#pragma unroll
    for (int r = 0; r < 8; ++r)
        hp[(size_t)r * XS] = c[r];
}

// ---------------- kernel 2: per-(node,head) attention logits ----------------
__global__ void calc_alpha(const float* __restrict__ h,
                           const float* __restrict__ a_s, const float* __restrict__ a_d,
                           float* __restrict__ as_, float* __restrict__ ad_,
                           int H, int C) {
    int t = blockIdx.x * blockDim.x + threadIdx.x;
    if (t >= N_NODES * H) return;
    const int n = t / H, hh = t - n * H;
    const float* hp = h + n * XS + hh * C;
    float s = 0.0f, d = 0.0f;
    for (int c = 0; c < C; ++c) {
        const float v = hp[c];
        s += v * a_s[hh * C + c];
        d += v * a_d[hh * C + c];
    }
    as_[n * 4 + hh] = s;
    ad_[n * 4 + hh] = d;
}

// ---------------- kernel 3: per-layer init (out=0, max=-inf, denom=0) ----------------
__global__ void init_layer(float* __restrict__ ob, float* __restrict__ mx,
                           float* __restrict__ dn, int HC, int H) {
    int n = blockIdx.x * blockDim.x + threadIdx.x;
    if (n >= N_NODES) return;
    for (int j = 0; j < HC; ++j) ob[n * XS + j] = 0.0f;
    for (int hh = 0; hh < H; ++hh) {
        mx[n * 4 + hh] = -__builtin_inff();
        dn[n * 4 + hh] = 0.0f;
    }
}

// ---------------- kernel 4: segment max over destinations ----------------
__global__ void edge_max(const int* __restrict__ ei, const float* __restrict__ as_,
                         const float* __restrict__ ad_, float* __restrict__ mx, int H) {
    int e = blockIdx.x * blockDim.x + threadIdx.x;
    if (e >= EP) return;
    int s, d; edge_nodes(ei, e, s, d);
    for (int hh = 0; hh < H; ++hh) {
        const float v = leaky(as_[s * 4 + hh] + ad_[d * 4 + hh]);
        atomicMaxF32(&mx[d * 4 + hh], v);
    }
}

// ---------------- kernel 5: segment sum of exp(e - max) ----------------
__global__ void edge_sum(const int* __restrict__ ei, const float* __restrict__ as_,
                         const float* __restrict__ ad_, const float* __restrict__ mx,
                         float* __restrict__ dn, int H) {
    int e = blockIdx.x * blockDim.x + threadIdx.x;
    if (e >= EP) return;
    int s, d; edge_nodes(ei, e, s, d);
    for (int hh = 0; hh < H; ++hh) {
        const float v  = leaky(as_[s * 4 + hh] + ad_[d * 4 + hh]);
        const float ex = expf(v - mx[d * 4 + hh]);
        atomicAdd(&dn[d * 4 + hh], ex);
    }
}

// ---------------- kernel 6: weighted message aggregation ----------------
__global__ void edge_agg(const int* __restrict__ ei, const float* __restrict__ as_,
                         const float* __restrict__ ad_, const float* __restrict__ mx,
                         const float* __restrict__ dn, const float* __restrict__ h,
                         float* __restrict__ ob, int H, int C) {
    int t = blockIdx.x * blockDim.x + threadIdx.x;
    if (t >= EP * H) return;
    const int e = t / H, hh = t - e * H;
    int s, d; edge_nodes(ei, e, s, d);
    const float v     = leaky(as_[s * 4 + hh] + ad_[d * 4 + hh]);
    const float alpha = expf(v - mx[d * 4 + hh]) / (dn[d * 4 + hh] + 1e-16f);
    const float* hp = h + s * XS + hh * C;
    float* op       = ob + d * XS + hh * C;
    for (int c = 0; c < C; ++c) atomicAdd(&op[c], hp[c] * alpha);
}

// ---------------- kernel 7: activations ----------------
__global__ void act_relu(const float* __restrict__ ob, const float* __restrict__ bias,
                         float* __restrict__ xb, int HC) {
    int t = blockIdx.x * blockDim.x + threadIdx.x;
    int n = t >> 7, j = t & 127;
    if (n >= N_NODES) return;
    float v = 0.0f;
    if (j < HC) {
        v = ob[n * XS + j] + bias[j];
        v = v > 0.0f ? v : 0.0f;
    }
    xb[n * XS + j] = v;    // zero-pad tail for next layer's K loop
}

__global__ void act_sigmoid_out(const float* __restrict__ ob, const float* __restrict__ bias,
                                float* __restrict__ out, int HC) {
    int t = blockIdx.x * blockDim.x + threadIdx.x;
    if (t >= N_NODES * HC) return;
    const int n = t / HC, j = t - n * HC;
    const float v = ob[n * XS + j] + bias[j];
    out[t] = 1.0f / (1.0f + expf(-v));
}

// ---------------- host driver ----------------
extern "C" void kernel_launch(void* const* d_in, const int* in_sizes, int n_in,
                              void* d_out, int out_size, void* d_ws, size_t ws_size,
                              hipStream_t stream) {
    (void)in_sizes; (void)n_in; (void)out_size; (void)ws_size;
    const int*   node_ids   = (const int*)d_in[0];
    const float* node_feats = (const float*)d_in[1];
    const int*   ei         = (const int*)d_in[2];
    /* d_in[3] = batch (unused) */
    const float* emb        = (const float*)d_in[4];
    const float* Wm[4] = { (const float*)d_in[5], (const float*)d_in[9],
                           (const float*)d_in[13], (const float*)d_in[17] };
    const float* As[4] = { (const float*)d_in[6], (const float*)d_in[10],
                           (const float*)d_in[14], (const float*)d_in[18] };
    const float* Ad[4] = { (const float*)d_in[7], (const float*)d_in[11],
                           (const float*)d_in[15], (const float*)d_in[19] };
    const float* Bi[4] = { (const float*)d_in[8], (const float*)d_in[12],
                           (const float*)d_in[16], (const float*)d_in[20] };
    const int K_[4] = {23, 128, 32, 128};
    const int H_[4] = {4, 1, 4, 1};
    const int C_[4] = {32, 32, 32, 16};

    // workspace carve-out (floats):  3 * N*128 + 4 * N*4 + 4096  ~= 80 MB
    float* xb  = (float*)d_ws;
    float* hb  = xb  + (size_t)N_NODES * XS;
    float* ob  = hb  + (size_t)N_NODES * XS;
    float* asb = ob  + (size_t)N_NODES * XS;
    float* adb = asb + (size_t)N_NODES * 4;
    float* mxb = adb + (size_t)N_NODES * 4;
    float* dnb = mxb + (size_t)N_NODES * 4;
    float* wpk = dnb + (size_t)N_NODES * 4;   // packed weights, <= 4096 floats

    build_x<<<cdiv(N_NODES * XS, 256), 256, 0, stream>>>(node_ids, node_feats, emb, xb);

    for (int l = 0; l < 4; ++l) {
        const int K = K_[l], H = H_[l], C = C_[l], HC = H * C;
        const int KB = ((K + 3) & ~3) / 4;           // K-steps of 4
        const int wtot = (HC / 16) * KB * 64;        // packed-W element count

        pack_w<<<cdiv(wtot, 256), 256, 0, stream>>>(Wm[l], wpk, K, HC, KB);
        gat_gemm<<<dim3(N_NODES / 16, HC / 16), 32, 0, stream>>>(xb, wpk, hb, KB);
        calc_alpha<<<cdiv(N_NODES * H, 256), 256, 0, stream>>>(hb, As[l], Ad[l], asb, adb, H, C);
        init_layer<<<cdiv(N_NODES, 256), 256, 0, stream>>>(ob, mxb, dnb, HC, H);
        edge_max<<<cdiv(EP, 256), 256, 0, stream>>>(ei, asb, adb, mxb, H);
        edge_sum<<<cdiv(EP, 256), 256, 0, stream>>>(ei, asb, adb, mxb, dnb, H);
        edge_agg<<<cdiv(EP * H, 256), 256, 0, stream>>>(ei, asb, adb, mxb, dnb, hb, ob, H, C);

        if (l < 3)
            act_relu<<<cdiv(N_NODES * XS, 256), 256, 0, stream>>>(ob, Bi[l], xb, HC);
        else
            act_sigmoid_out<<<cdiv(N_NODES * 16, 256), 256, 0, stream>>>(ob, Bi[l], (float*)d_out, 16);
    }
}